// SGModel_37666863186543
// MI455X (gfx1250) — compile-verified
//
#include <hip/hip_runtime.h>

typedef float v2f __attribute__((ext_vector_type(2)));
typedef float v8f __attribute__((ext_vector_type(8)));

#define C 64
#define PS 160            // padded float stride per k-pair row (128 data + 32 pad banks)
#define WLDS_FLOATS (32 * PS)   // 32 k-pairs

// ---------------------------------------------------------------------------
// Degree accumulation: deg[dst] += 1.0 per edge (f32 exact for deg < 2^24)
// ---------------------------------------------------------------------------
__global__ void sg_deg_kernel(const int* __restrict__ dst,
                              float* __restrict__ deg, int n_edges) {
    int e = blockIdx.x * blockDim.x + threadIdx.x;
    if (e < n_edges) atomicAdd(&deg[dst[e]], 1.0f);
}

// norm[v] = max(deg,1)^-0.5   (in place)
__global__ void sg_norm_kernel(float* __restrict__ norm, int n) {
    int v = blockIdx.x * blockDim.x + threadIdx.x;
    if (v < n) {
        float d = fmaxf(norm[v], 1.0f);
        norm[v] = 1.0f / sqrtf(d);
    }
}

// ---------------------------------------------------------------------------
// Edge scatter: agg[dst][c] += x[src][c] * norm[src]
// 16 threads per edge, float4 per thread -> coalesced loads & atomics
// ---------------------------------------------------------------------------
__global__ void sg_scatter_kernel(const float* __restrict__ x,
                                  const float* __restrict__ norm,
                                  const int* __restrict__ src,
                                  const int* __restrict__ dst,
                                  float* __restrict__ agg, int n_edges) {
    long tid = (long)blockIdx.x * blockDim.x + threadIdx.x;
    long total = (long)n_edges * 16;
    if (tid >= total) return;
    int e  = (int)(tid >> 4);
    int c4 = (int)(tid & 15) * 4;
    int s = src[e];
    int d = dst[e];
    float nv = norm[s];
    float4 v = *(const float4*)(x + (size_t)s * C + c4);
    float* ap = agg + (size_t)d * C + c4;
    atomicAdd(ap + 0, v.x * nv);
    atomicAdd(ap + 1, v.y * nv);
    atomicAdd(ap + 2, v.z * nv);
    atomicAdd(ap + 3, v.w * nv);
}

// ---------------------------------------------------------------------------
// LDS staging: W[k][col] -> Wlds[(k>>1)*PS + 2*col + (k&1)]
// Packs K-adjacent pairs contiguously so each B fragment is one ds_load_b64;
// PS=160 keeps the two half-wave kphases on disjoint 32-bank halves.
// ---------------------------------------------------------------------------
__device__ __forceinline__ void stage_w_paired(const float* __restrict__ W,
                                               float* __restrict__ Wlds) {
    for (int i = threadIdx.x; i < C * C; i += blockDim.x) {
        int k = i >> 6;
        int col = i & 63;
        Wlds[(k >> 1) * PS + 2 * col + (k & 1)] = W[i];
    }
}

// ---------------------------------------------------------------------------
// Residual GEMM (layer 0 only): out = x @ Wres + bres     (fp32 WMMA 16x16x4)
// One wave computes a 16x64 output stripe; 8 waves / block.
// ---------------------------------------------------------------------------
__global__ void sg_res_gemm_kernel(const float* __restrict__ xin,
                                   const float* __restrict__ Wres,
                                   const float* __restrict__ bres,
                                   float* __restrict__ out, int n_nodes) {
    __shared__ float Wlds[WLDS_FLOATS];
    __shared__ float blds[C];

    stage_w_paired(Wres, Wlds);
    if (threadIdx.x < C) blds[threadIdx.x] = bres[threadIdx.x];
    __syncthreads();

    int wave     = (int)((blockIdx.x * blockDim.x + threadIdx.x) >> 5);
    int row_base = wave * 16;
    if (row_base >= n_nodes) return;          // wave-uniform

    int lane   = threadIdx.x & 31;
    int r      = lane & 15;
    int kphase = lane >> 4;

    const float* xRow = xin + (size_t)(row_base + r) * C;

    v8f acc[4] = {};
#pragma unroll
    for (int kk = 0; kk < C; kk += 4) {
        int kOff    = kk + 2 * kphase;        // even -> aligned b64
        int pairRow = (kk >> 1) + kphase;
        v2f a = *(const v2f*)(xRow + kOff);
#pragma unroll
        for (int nt = 0; nt < 4; ++nt) {
            int col = nt * 16 + r;
            v2f bb = *(const v2f*)(Wlds + pairRow * PS + 2 * col);
            acc[nt] = __builtin_amdgcn_wmma_f32_16x16x4_f32(
                false, a, false, bb, (short)0, acc[nt], false, false);
        }
    }

#pragma unroll
    for (int nt = 0; nt < 4; ++nt) {
        int col = nt * 16 + r;
#pragma unroll
        for (int m = 0; m < 8; ++m) {
            int row = row_base + m + 8 * kphase;
            out[(size_t)row * C + col] = acc[nt][m] + blds[col];
        }
    }
}

// ---------------------------------------------------------------------------
// Main GEMM + epilogue:  out = relu((agg*norm) @ W + b) + res[row][col]
// `res` always aliases `out` in this pipeline (same-lane read-then-write, safe).
// ---------------------------------------------------------------------------
__global__ void sg_main_gemm_kernel(const float* __restrict__ agg,
                                    const float* __restrict__ norm,
                                    const float* __restrict__ W,
                                    const float* __restrict__ bias,
                                    const float* __restrict__ res,
                                    float* __restrict__ out, int n_nodes) {
    __shared__ float Wlds[WLDS_FLOATS];
    __shared__ float blds[C];

    stage_w_paired(W, Wlds);
    if (threadIdx.x < C) blds[threadIdx.x] = bias[threadIdx.x];
    __syncthreads();

    int wave     = (int)((blockIdx.x * blockDim.x + threadIdx.x) >> 5);
    int row_base = wave * 16;
    if (row_base >= n_nodes) return;          // wave-uniform

    int lane   = threadIdx.x & 31;
    int r      = lane & 15;
    int kphase = lane >> 4;

    int   arow  = row_base + r;
    float normv = norm[arow];
    const float* aRow = agg + (size_t)arow * C;

    v8f acc[4] = {};
#pragma unroll
    for (int kk = 0; kk < C; kk += 4) {
        int kOff    = kk + 2 * kphase;
        int pairRow = (kk >> 1) + kphase;
        v2f a = *(const v2f*)(aRow + kOff);
        a.x *= normv;
        a.y *= normv;
#pragma unroll
        for (int nt = 0; nt < 4; ++nt) {
            int col = nt * 16 + r;
            v2f bb = *(const v2f*)(Wlds + pairRow * PS + 2 * col);
            acc[nt] = __builtin_amdgcn_wmma_f32_16x16x4_f32(
                false, a, false, bb, (short)0, acc[nt], false, false);
        }
    }

#pragma unroll
    for (int nt = 0; nt < 4; ++nt) {
        int col = nt * 16 + r;
#pragma unroll
        for (int m = 0; m < 8; ++m) {
            int row = row_base + m + 8 * kphase;
            float h = acc[nt][m] + blds[col];
            float rv = res[(size_t)row * C + col];
            out[(size_t)row * C + col] = fmaxf(h, 0.0f) + rv;
        }
    }
}

// ---------------------------------------------------------------------------
extern "C" void kernel_launch(void* const* d_in, const int* in_sizes, int n_in,
                              void* d_out, int out_size, void* d_ws, size_t ws_size,
                              hipStream_t stream) {
    const float* features = (const float*)d_in[0];   // [N,64]
    const float* W        = (const float*)d_in[1];   // [3,64,64]
    const float* b        = (const float*)d_in[2];   // [3,64]
    const float* Wres     = (const float*)d_in[3];   // [64,64]
    const float* bres     = (const float*)d_in[4];   // [64]
    const int*   src      = (const int*)d_in[5];     // [E]
    const int*   dst      = (const int*)d_in[6];     // [E]
    float*       out      = (float*)d_out;           // [N,64], also inter-layer x / res

    const int n_nodes = in_sizes[0] / C;
    const int n_edges = in_sizes[5];

    // Workspace: norm [N] f32, then agg [N,64] f32 (256B aligned)
    float* normBuf = (float*)d_ws;
    size_t normBytes = (((size_t)n_nodes * sizeof(float)) + 255) & ~(size_t)255;
    float* agg = (float*)((char*)d_ws + normBytes);
    size_t aggBytes = (size_t)n_nodes * C * sizeof(float);

    // Degrees -> symmetric norm
    hipMemsetAsync(normBuf, 0, (size_t)n_nodes * sizeof(float), stream);
    sg_deg_kernel<<<(n_edges + 255) / 256, 256, 0, stream>>>(dst, normBuf, n_edges);
    sg_norm_kernel<<<(n_nodes + 255) / 256, 256, 0, stream>>>(normBuf, n_nodes);

    const int waves   = (n_nodes + 15) / 16;
    const int gblocks = (waves + 7) / 8;            // 8 waves (256 thr) / block
    const long stot   = (long)n_edges * 16;
    const int sblocks = (int)((stot + 255) / 256);

    for (int layer = 0; layer < 3; ++layer) {
        const float* xin = (layer == 0) ? features : out;

        hipMemsetAsync(agg, 0, aggBytes, stream);
        sg_scatter_kernel<<<sblocks, 256, 0, stream>>>(xin, normBuf, src, dst,
                                                       agg, n_edges);
        if (layer == 0) {
            // Residual pass: out = features @ Wres + bres
            sg_res_gemm_kernel<<<gblocks, 256, 0, stream>>>(features, Wres, bres,
                                                            out, n_nodes);
        }
        // Main pass: out = relu((agg*norm)@W + b) + out
        sg_main_gemm_kernel<<<gblocks, 256, 0, stream>>>(
            agg, normBuf, W + (size_t)layer * C * C, b + (size_t)layer * C,
            out, out, n_nodes);
    }
}